// ChebTimeConvDeprecated_13288628814255
// MI455X (gfx1250) — compile-verified
//
#include <hip/hip_runtime.h>
#include <hip/hip_bf16.h>

typedef __attribute__((ext_vector_type(2))) float v2f;
typedef __attribute__((ext_vector_type(8))) float v8f;

// Problem constants (fixed by the reference setup)
#define QQ 2
#define NN 50000
#define HH 4
#define FF 32
#define GG 32
#define KK 4
#define EE 800000
#define CHF (HH * FF)          // 128 features per (q,n)
#define QN (QQ * NN)           // 100000 GEMM rows
#define TILES (QN / 16)        // 6250 row tiles (exact)
#define WSTRIDE 132            // padded LDS row stride (words): bank = (4g+kk)%64, conflict-free

// ---------- small utility kernels ----------

__global__ void zero_f32(float* __restrict__ p, int n) {
  int i = blockIdx.x * blockDim.x + threadIdx.x;
  if (i < n) p[i] = 0.0f;
}

__global__ void neg_copy_f32(const float* __restrict__ src, float* __restrict__ dst, int n) {
  int i = blockIdx.x * blockDim.x + threadIdx.x;
  if (i < n) dst[i] = -src[i];
}

__global__ void neg_inplace_f32(float* __restrict__ p, int n) {
  int i = blockIdx.x * blockDim.x + threadIdx.x;
  if (i < n) p[i] = -p[i];
}

__global__ void init_out_bias(const float* __restrict__ bias, float* __restrict__ out, int n) {
  int i = blockIdx.x * blockDim.x + threadIdx.x;
  if (i < n) out[i] = bias[i & (GG - 1)];
}

// ---------- graph preprocessing ----------

__global__ void degree_kernel(const int* __restrict__ row, const int* __restrict__ col,
                              float* __restrict__ deg, int E) {
  int e = blockIdx.x * blockDim.x + threadIdx.x;
  if (e >= E) return;
  int r = row[e];
  if (r != col[e]) atomicAdd(&deg[r], 1.0f);
}

__global__ void lap_kernel(const int* __restrict__ row, const int* __restrict__ col,
                           const float* __restrict__ ew, const float* __restrict__ deg,
                           float* __restrict__ lap, int E) {
  int e = blockIdx.x * blockDim.x + threadIdx.x;
  if (e >= E) return;
  int r = row[e], c = col[e];
  float l = 0.0f;
  if (r != c) {
    float dr = deg[r], dc = deg[c];
    float ir = (dr > 0.0f) ? rsqrtf(dr) : 0.0f;
    float ic = (dc > 0.0f) ? rsqrtf(dc) : 0.0f;
    l = -ir * ew[e] * ic;
  }
  lap[e] = l;
}

// ---------- SpMM: Tout[q, row, :] += scale * lap[e] * Tin[q, col, :] ----------
// 64 threads per edge (2 q-planes x 32 float4 slots of the 128-float node block).
// Atomic/L2-bound: Tx tensors (51.2 MB) are L2-resident on MI455X (192 MB).
__global__ void spmm_kernel(const float* __restrict__ Tin, float* __restrict__ Tout,
                            const float* __restrict__ lap,
                            const int* __restrict__ row, const int* __restrict__ col,
                            float scale, int E) {
  int gid = blockIdx.x * blockDim.x + threadIdx.x;
  if (gid >= E * 64) return;
  int e = gid >> 6;
  float l = lap[e] * scale;
  if (l == 0.0f) return;                       // masked / zero-degree edge (wave-uniform)
  int c4 = gid & 63;                           // which float4 of the 256 channels
  int q = c4 >> 5;
  int f4 = (c4 & 31) << 2;
  int r = row[e], c = col[e];
  size_t inIdx  = ((size_t)q * NN + c) * CHF + f4;
  size_t outIdx = ((size_t)q * NN + r) * CHF + f4;
  const float4 v = *(const float4*)(Tin + inIdx);
  atomicAdd(Tout + outIdx + 0, l * v.x);
  atomicAdd(Tout + outIdx + 1, l * v.y);
  atomicAdd(Tout + outIdx + 2, l * v.z);
  atomicAdd(Tout + outIdx + 3, l * v.w);
}

// ---------- GEMM-accumulate via V_WMMA_F32_16X16X4_F32 ----------
// out[row, g] += sum_f T[row, f] * Wk[f, g];   T: [QN, 128], Wk: [128, 32], out: [QN, 32]
// One wave owns a full 16x32 output tile (two 16x16 accumulators sharing one A fetch);
// 8 waves/block -> 8 row tiles per block. Wk staged TRANSPOSED in LDS so each B operand
// pair (K-adjacent) is one aligned ds_load_b64 straight into a WMMA register pair.
__global__ __launch_bounds__(256) void gemm_wmma_acc(const float* __restrict__ T,
                                                     const float* __restrict__ Wk,
                                                     float* __restrict__ out) {
  __shared__ float sWt[GG * WSTRIDE];              // W^T [g][f], padded rows: ~16.9 KB

  // stage transposed: consecutive threads vary f -> conflict-free LDS writes
#pragma unroll
  for (int i = 0; i < 16; ++i) {
    int idx = threadIdx.x + 256 * i;               // 0..4095
    int f = idx & (CHF - 1);
    int g = idx >> 7;
    sWt[g * WSTRIDE + f] = Wk[(size_t)f * GG + g];
  }
  __syncthreads();

  const int wave = threadIdx.x >> 5;
  const int lane = threadIdx.x & 31;
  const int tile = blockIdx.x * 8 + wave;          // row tile index (wave-uniform)

  if (tile < TILES) {                              // wave-uniform -> EXEC all-ones at WMMA
    const int laneLo = lane & 15;
    const int hi     = lane >> 4;                  // lanes 16-31 carry K+2 / M+8 halves
    const int r0     = tile << 4;

    const float* __restrict__ Arow = T + (size_t)(r0 + laneLo) * CHF;
    const float* __restrict__ Bp0  = sWt + laneLo * WSTRIDE;          // g = laneLo
    const float* __restrict__ Bp1  = sWt + (laneLo + 16) * WSTRIDE;   // g = laneLo+16

    v8f acc0 = {};                                 // G columns 0..15
    v8f acc1 = {};                                 // G columns 16..31
#pragma unroll
    for (int k0 = 0; k0 < CHF; k0 += 4) {
      const int kk = k0 + 2 * hi;
      // A 16x4: lane<16 -> A[M=laneLo][k0..k0+1]; lane>=16 -> K+2 pair (b64)
      v2f a = *(const v2f*)(Arow + kk);
      // B 4x16 pairs: contiguous in transposed LDS -> single aligned ds_load_b64 each
      v2f b0 = *(const v2f*)(Bp0 + kk);
      v2f b1 = *(const v2f*)(Bp1 + kk);
      acc0 = __builtin_amdgcn_wmma_f32_16x16x4_f32(false, a, false, b0, (short)0, acc0,
                                                   false, false);
      acc1 = __builtin_amdgcn_wmma_f32_16x16x4_f32(false, a, false, b1, (short)0, acc1,
                                                   false, false);
    }

    // C/D layout: VGPR i holds M = i + 8*hi, N = laneLo (+16 for acc1)
#pragma unroll
    for (int i = 0; i < 8; ++i) {
      const size_t o = (size_t)(r0 + i + 8 * hi) * GG + laneLo;
      out[o] += acc0[i];
      out[o + 16] += acc1[i];
    }
  }
}

// ---------- orchestration ----------

extern "C" void kernel_launch(void* const* d_in, const int* in_sizes, int n_in,
                              void* d_out, int out_size, void* d_ws, size_t ws_size,
                              hipStream_t stream) {
  const float* x      = (const float*)d_in[0];      // [Q, N, H, F] == [QN, 128]
  const int*   eidx   = (const int*)d_in[1];        // [2, E]
  const float* ew     = (const float*)d_in[2];      // [E]
  const float* weight = (const float*)d_in[3];      // [K, H, F, G]
  const float* bias   = (const float*)d_in[4];      // [G]
  float*       out    = (float*)d_out;              // [Q, N, G]

  const int* row = eidx;
  const int* col = eidx + EE;

  // workspace layout (floats): deg[N] | lap[E] | A[QN*128] | B[QN*128]  (~106 MB)
  float* ws  = (float*)d_ws;
  float* deg = ws;
  float* lap = deg + NN;
  float* A   = lap + EE;
  float* B   = A + (size_t)QN * CHF;

  const int TXN   = QN * CHF;                       // 12.8M floats per Tx buffer
  const int OUTN  = QN * GG;                        // 3.2M output floats
  const int SPMMT = EE * 64;                        // spmm thread count

  const int TB = 256;
  dim3 blk(TB);
  const int gTx   = (TXN + TB - 1) / TB;
  const int gOut  = (OUTN + TB - 1) / TB;
  const int gE    = (EE + TB - 1) / TB;
  const int gSp   = (SPMMT + TB - 1) / TB;
  const int gGemm = (TILES + 7) / 8;                // 8 row tiles per 8-wave block

  const size_t wStride = (size_t)HH * FF * GG;      // 4096 floats per k-slice

  // 1) degree + laplacian edge values
  zero_f32<<<(NN + TB - 1) / TB, blk, 0, stream>>>(deg, NN);
  degree_kernel<<<gE, blk, 0, stream>>>(row, col, deg, EE);
  lap_kernel<<<gE, blk, 0, stream>>>(row, col, ew, deg, lap, EE);

  // 2) out = bias; out += Tx0 @ W0   (Tx0 = x)
  init_out_bias<<<gOut, blk, 0, stream>>>(bias, out, OUTN);
  gemm_wmma_acc<<<gGemm, blk, 0, stream>>>(x, weight + 0 * wStride, out);

  // 3) Tx1 = spmm(x) in A; out += Tx1 @ W1
  zero_f32<<<gTx, blk, 0, stream>>>(A, TXN);
  spmm_kernel<<<gSp, blk, 0, stream>>>(x, A, lap, row, col, 1.0f, EE);
  gemm_wmma_acc<<<gGemm, blk, 0, stream>>>(A, weight + 1 * wStride, out);

  // 4) Tx2 = 2*spmm(Tx1) - Tx0 in B (init B = -x, accumulate 2*lap); out += Tx2 @ W2
  neg_copy_f32<<<gTx, blk, 0, stream>>>(x, B, TXN);
  spmm_kernel<<<gSp, blk, 0, stream>>>(A, B, lap, row, col, 2.0f, EE);
  gemm_wmma_acc<<<gGemm, blk, 0, stream>>>(B, weight + 2 * wStride, out);

  // 5) Tx3 = 2*spmm(Tx2) - Tx1 in A (init A = -Tx1 in place); out += Tx3 @ W3
  neg_inplace_f32<<<gTx, blk, 0, stream>>>(A, TXN);
  spmm_kernel<<<gSp, blk, 0, stream>>>(B, A, lap, row, col, 2.0f, EE);
  gemm_wmma_acc<<<gGemm, blk, 0, stream>>>(A, weight + 3 * wStride, out);
}